// GCMCGraphConv_77300821393408
// MI455X (gfx1250) — compile-verified
//
#include <hip/hip_runtime.h>

typedef float v2f __attribute__((ext_vector_type(2)));
typedef float v8f __attribute__((ext_vector_type(8)));

// ---------------------------------------------------------------- zero d_out
__global__ void gcmc_zero(float* __restrict__ out, int n) {
  int i = blockIdx.x * blockDim.x + threadIdx.x;
  int stride = gridDim.x * blockDim.x;
  for (; i < n; i += stride) out[i] = 0.0f;
}

// ------------------------------------------------------- final h * ci scale
__global__ void gcmc_scale(float* __restrict__ out, const float* __restrict__ ci, int n) {
  int i = blockIdx.x * blockDim.x + threadIdx.x;
  int stride = gridDim.x * blockDim.x;
  for (; i < n; i += stride) out[i] *= ci[i >> 6];   // D == 64
}

// ----------------------------------------------------------- main edge pass
// Per wave: 16 edges. rf tile (16x64) + extras tile (pa/ra dots) via
// V_WMMA_F32_16X16X4_F32 chains; then fuse gather/combine/scatter.
__launch_bounds__(256)
__global__ void gcmc_main(const float* __restrict__ weight,
                          const float* __restrict__ review_feat,
                          const float* __restrict__ attn,
                          const float* __restrict__ cj,
                          const float* __restrict__ prob_w,
                          const float* __restrict__ rsw,
                          const float* __restrict__ review_w,
                          const int*   __restrict__ src_idx,
                          const int*   __restrict__ dst_idx,
                          float* __restrict__ out,
                          int E, int ntiles) {
  // B fragments in native WMMA B-layout:
  // fragment (nt, kk): 32 lanes x 2 floats; lane = {hi: K-pair select, n: column}
  // nt 0..3 = review_w^T column tiles, nt 4 = [prob_w | review_score_w | 0...]
  __shared__ float ldsB[5 * 16 * 64];   // 20 KB

  const int tid = threadIdx.x;
  for (int s = tid; s < 5 * 16 * 32; s += 256) {
    int ln = s & 31;
    int kk = (s >> 5) & 15;
    int nt = s >> 9;
    int hb = ln >> 4;
    int nc = ln & 15;
    int k0 = 4 * kk + 2 * hb;
    float v0, v1;
    if (nt < 4) {
      int col = nt * 16 + nc;                 // B[k][n] = review_w[n][k]
      v0 = review_w[col * 64 + k0];
      v1 = review_w[col * 64 + k0 + 1];
    } else if (nc == 0) { v0 = prob_w[k0]; v1 = prob_w[k0 + 1]; }
    else if (nc == 1)   { v0 = rsw[k0];    v1 = rsw[k0 + 1];    }
    else                { v0 = 0.0f;       v1 = 0.0f;           }
    ldsB[2 * s]     = v0;
    ldsB[2 * s + 1] = v1;
  }
  __syncthreads();

  const int lane = tid & 31;
  const int hi   = lane >> 4;          // half-wave select
  const int n    = lane & 15;          // column within 16-wide tile / row id
  const int hi8  = hi * 8;

  const int waveId     = blockIdx.x * 8 + (tid >> 5);
  const int waveStride = gridDim.x * 8;

  for (int tile = waveId; tile < ntiles; tile += waveStride) {
    const int base = tile * 16;

    // per-row scalars: lanes 0..15 and 16..31 both hold rows 0..15
    int er = base + n; if (er >= E) er = E - 1;
    float attn_l = attn[er];
    int   s_l    = src_idx[er];
    int   d_l    = dst_idx[er];
    float cj_l   = cj[s_l];

    // A-matrix source: lane holds row (base + n), K-pair selected by hi
    const float* fr = review_feat + (size_t)er * 64 + 2 * hi;

    v8f acc0 = {}, acc1 = {}, acc2 = {}, acc3 = {}, acc4 = {};
    #pragma unroll
    for (int kk = 0; kk < 16; ++kk) {
      v2f a = *(const v2f*)(fr + 4 * kk);
      const float* bb = ldsB + kk * 64 + 2 * lane;
      v2f b0 = *(const v2f*)(bb);
      v2f b1 = *(const v2f*)(bb + 1024);
      v2f b2 = *(const v2f*)(bb + 2048);
      v2f b3 = *(const v2f*)(bb + 3072);
      v2f b4 = *(const v2f*)(bb + 4096);
      acc0 = __builtin_amdgcn_wmma_f32_16x16x4_f32(false, a, false, b0, (short)0, acc0, false, false);
      acc1 = __builtin_amdgcn_wmma_f32_16x16x4_f32(false, a, false, b1, (short)0, acc1, false, false);
      acc2 = __builtin_amdgcn_wmma_f32_16x16x4_f32(false, a, false, b2, (short)0, acc2, false, false);
      acc3 = __builtin_amdgcn_wmma_f32_16x16x4_f32(false, a, false, b3, (short)0, acc3, false, false);
      acc4 = __builtin_amdgcn_wmma_f32_16x16x4_f32(false, a, false, b4, (short)0, acc4, false, false);
    }

    // C/D layout: lane L, VGPR k -> element (M = k + 8*(L>=16), N = L%16).
    // extras tile acc4: col 0 = prob dot, col 1 = review_score dot.
    #pragma unroll
    for (int k = 0; k < 8; ++k) {
      float sig  = 1.0f / (1.0f + __expf(-acc4[k]));        // sigmoid everywhere, pick lanes below
      float pa_k = __shfl(sig, lane & 16, 32);              // col 0 of this half's rows
      float ra_k = __shfl(sig, (lane & 16) | 1, 32);        // col 1
      int   rsl  = k + hi8;                                 // row within tile for this lane
      float att_k = __shfl(attn_l, rsl, 32);
      float cj_k  = __shfl(cj_l,  rsl, 32);
      int   sr    = __shfl(s_l,   rsl, 32);
      int   dr    = __shfl(d_l,   rsl, 32);
      float raa   = ra_k * att_k;

      if (base + rsl < E) {
        const float* wrow = weight + ((size_t)sr << 6) + n;
        float*       orow = out    + ((size_t)dr << 6) + n;
        float v0 = fmaf(wrow[0],  pa_k, acc0[k] * raa) * cj_k;
        float v1 = fmaf(wrow[16], pa_k, acc1[k] * raa) * cj_k;
        float v2 = fmaf(wrow[32], pa_k, acc2[k] * raa) * cj_k;
        float v3 = fmaf(wrow[48], pa_k, acc3[k] * raa) * cj_k;
        atomicAdd(orow +  0, v0);
        atomicAdd(orow + 16, v1);
        atomicAdd(orow + 32, v2);
        atomicAdd(orow + 48, v3);
      }
    }
  }
}

extern "C" void kernel_launch(void* const* d_in, const int* in_sizes, int n_in,
                              void* d_out, int out_size, void* d_ws, size_t ws_size,
                              hipStream_t stream) {
  const float* weight      = (const float*)d_in[0];
  const float* review_feat = (const float*)d_in[1];
  const float* attn        = (const float*)d_in[2];
  const float* cj          = (const float*)d_in[3];
  const float* ci          = (const float*)d_in[4];
  const float* prob_w      = (const float*)d_in[5];
  const float* rsw         = (const float*)d_in[6];
  const float* review_w    = (const float*)d_in[7];
  const int*   src_idx     = (const int*)d_in[8];
  const int*   dst_idx     = (const int*)d_in[9];
  float* out = (float*)d_out;

  const int E      = in_sizes[2];          // attn is (E,)
  const int ntiles = (E + 15) / 16;

  int zblocks = (out_size + 255) / 256;
  if (zblocks > 4096) zblocks = 4096;
  gcmc_zero<<<zblocks, 256, 0, stream>>>(out, out_size);

  int gblocks = (ntiles + 7) / 8;
  if (gblocks > 2048) gblocks = 2048;
  gcmc_main<<<gblocks, 256, 0, stream>>>(weight, review_feat, attn, cj, prob_w,
                                         rsw, review_w, src_idx, dst_idx, out,
                                         E, ntiles);

  gcmc_scale<<<zblocks, 256, 0, stream>>>(out, ci, out_size);
}